// GCN_15522011808374
// MI455X (gfx1250) — compile-verified
//
#include <hip/hip_runtime.h>
#include <math.h>

#define N_NODES 100000
#define N_EDGES 800000
#define DIM     128
#define N_GRAPHS 512

typedef __attribute__((ext_vector_type(2))) float v2f;
typedef __attribute__((ext_vector_type(8))) float v8f;

// ---------------------------------------------------------------- zero fill
__global__ void zero_kernel(float* __restrict__ p, int n) {
    int i = blockIdx.x * blockDim.x + threadIdx.x;
    if (i < n) p[i] = 0.0f;
}

// ---------------------------------------------------------------- degrees
__global__ void degree_kernel(const int* __restrict__ src,
                              const int* __restrict__ dst,
                              float* __restrict__ deg_out,
                              float* __restrict__ deg_in) {
    int e = blockIdx.x * blockDim.x + threadIdx.x;
    if (e < N_EDGES) {
        atomicAdd(&deg_out[src[e]], 1.0f);
        atomicAdd(&deg_in[dst[e]], 1.0f);
    }
}

// deg -> rsqrt(max(deg,1)) in place (covers both deg_out and deg_in, contiguous)
__global__ void norm_kernel(float* __restrict__ deg, int n) {
    int i = blockIdx.x * blockDim.x + threadIdx.x;
    if (i < n) deg[i] = rsqrtf(fmaxf(deg[i], 1.0f));
}

// ---------------------------------------------------------------- edge scatter
// One wave (32 lanes) per edge; each lane handles 4 consecutive features.
__global__ void aggregate_kernel(const float* __restrict__ h,
                                 const float* __restrict__ norm_src,
                                 const int* __restrict__ src,
                                 const int* __restrict__ dst,
                                 float* __restrict__ agg) {
    int tid  = blockIdx.x * 256 + threadIdx.x;
    int e    = tid >> 5;
    int lane = tid & 31;
    if (e >= N_EDGES) return;
    int s = src[e];
    int d = dst[e];
    float ns = norm_src[s];
    const float4 v = *(const float4*)(h + (size_t)s * DIM + lane * 4);
    float* o = agg + (size_t)d * DIM + lane * 4;
    atomicAdd(o + 0, v.x * ns);
    atomicAdd(o + 1, v.y * ns);
    atomicAdd(o + 2, v.z * ns);
    atomicAdd(o + 3, v.w * ns);
}

// ---------------------------------------------------------------- WMMA GEMM
// out[row, col] = relu( (agg[row,:] * norm_dst[row]) @ W[:, col] + bias[col] )
// Wave-level: 16-row tile x all 128 cols (8 col tiles), K=128 via 32 steps of
// V_WMMA_F32_16X16X4_F32. W staged in LDS. 2 waves / block, 3125 blocks.
__launch_bounds__(64)
__global__ void gemm_bias_relu_kernel(const float* __restrict__ agg,
                                      const float* __restrict__ norm_dst,
                                      const float* __restrict__ W,
                                      const float* __restrict__ bias,
                                      float* __restrict__ out) {
    __shared__ float ldsW[DIM * DIM]; // 64 KB (320 KB/WGP available)

    // cooperative load of W into LDS (float4, coalesced)
    for (int i = threadIdx.x; i < DIM * DIM / 4; i += 64) {
        *(float4*)&ldsW[i * 4] = *(const float4*)&W[i * 4];
    }
    __syncthreads();

    const int wave = threadIdx.x >> 5;
    const int lane = threadIdx.x & 31;
    const int half = lane >> 4;   // 0: lanes 0-15, 1: lanes 16-31
    const int l16  = lane & 15;
    const int row_base = (blockIdx.x * 2 + wave) * 16;

    // A-fragment lane (half,l16) holds row M=l16 -> its norm_dst scale
    const float nd = norm_dst[row_base + l16];
    const float* arow = agg + (size_t)(row_base + l16) * DIM;

    v8f acc[8] = {};

    for (int k = 0; k < 32; ++k) {
        const int k0 = 4 * k + half * 2; // ISA A layout: V0->K=0/2, V1->K=1/3
        v2f a = *(const v2f*)(arow + k0);
        a.x *= nd;
        a.y *= nd;
#pragma unroll
        for (int c = 0; c < 8; ++c) {
            v2f b;
            b.x = ldsW[(k0 + 0) * DIM + c * 16 + l16]; // B row K=k0,   N=l16
            b.y = ldsW[(k0 + 1) * DIM + c * 16 + l16]; // B row K=k0+1, N=l16
            acc[c] = __builtin_amdgcn_wmma_f32_16x16x4_f32(
                false, a, false, b, (short)0, acc[c], false, false);
        }
    }

#pragma unroll
    for (int c = 0; c < 8; ++c) {
        const int col = c * 16 + l16;
        const float bv = bias[col];
#pragma unroll
        for (int r = 0; r < 8; ++r) {
            // C/D layout: VGPR r -> M=r (lanes 0-15), M=r+8 (lanes 16-31)
            const int row = row_base + r + half * 8;
            float v = acc[c][r] + bv;
            out[(size_t)row * DIM + col] = v > 0.0f ? v : 0.0f;
        }
    }
}

// ---------------------------------------------------------------- gate + pool
// One wave per node: gate = sigmoid(h[node,:] . wg + bg), then
// atomicAdd gate*h into wh[graph_ids[node], :].
__global__ void gate_pool_kernel(const float* __restrict__ h,
                                 const float* __restrict__ wg,
                                 const float* __restrict__ bg,
                                 const int* __restrict__ graph_ids,
                                 float* __restrict__ wh) {
    int tid  = blockIdx.x * 256 + threadIdx.x;
    int node = tid >> 5;
    int lane = tid & 31;
    if (node >= N_NODES) return;

    const float4 hv = *(const float4*)(h + (size_t)node * DIM + lane * 4);
    const float4 wv = *(const float4*)(wg + lane * 4);
    float p = hv.x * wv.x + hv.y * wv.y + hv.z * wv.z + hv.w * wv.w;
#pragma unroll
    for (int off = 16; off > 0; off >>= 1)
        p += __shfl_xor(p, off, 32);   // wave32 reduction; all lanes get sum

    const float gate = 1.0f / (1.0f + expf(-(p + bg[0])));
    const int g = graph_ids[node];
    float* o = wh + (size_t)g * DIM + lane * 4;
    atomicAdd(o + 0, gate * hv.x);
    atomicAdd(o + 1, gate * hv.y);
    atomicAdd(o + 2, gate * hv.z);
    atomicAdd(o + 3, gate * hv.w);
}

// ---------------------------------------------------------------- launcher
extern "C" void kernel_launch(void* const* d_in, const int* in_sizes, int n_in,
                              void* d_out, int out_size, void* d_ws, size_t ws_size,
                              hipStream_t stream) {
    const int*   src  = (const int*)d_in[0];
    const int*   dst  = (const int*)d_in[1];
    const int*   gid  = (const int*)d_in[2];
    const float* feat = (const float*)d_in[3];
    const float* w1   = (const float*)d_in[4];
    const float* b1   = (const float*)d_in[5];
    const float* w2   = (const float*)d_in[6];
    const float* b2   = (const float*)d_in[7];
    const float* wg   = (const float*)d_in[8];
    const float* bg   = (const float*)d_in[9];

    float* out  = (float*)d_out;
    float* wh   = out;                             // [512,128]
    float* hout = out + (size_t)N_GRAPHS * DIM;    // [100000,128]

    float* ws       = (float*)d_ws;
    float* norm_src = ws;                          // [N]      (deg_out -> norm)
    float* norm_dst = ws + N_NODES;                // [N]      (deg_in  -> norm)
    float* agg      = ws + 2 * N_NODES;            // [N,128]  (reused both layers)
    float* h1       = agg + (size_t)N_NODES * DIM; // [N,128]

    const int AGG_ELEMS = N_NODES * DIM;           // 12.8M

    // zero-init accumulation targets (d_out/d_ws are poisoned by harness)
    zero_kernel<<<(2 * N_NODES + 255) / 256, 256, 0, stream>>>(norm_src, 2 * N_NODES);
    zero_kernel<<<(AGG_ELEMS + 255) / 256, 256, 0, stream>>>(agg, AGG_ELEMS);
    zero_kernel<<<(N_GRAPHS * DIM + 255) / 256, 256, 0, stream>>>(wh, N_GRAPHS * DIM);

    // degrees + normalization factors
    degree_kernel<<<(N_EDGES + 255) / 256, 256, 0, stream>>>(src, dst, norm_src, norm_dst);
    norm_kernel<<<(2 * N_NODES + 255) / 256, 256, 0, stream>>>(norm_src, 2 * N_NODES);

    const int agg_blocks  = (N_EDGES * 32) / 256;  // 100000, exact
    const int gemm_blocks = N_NODES / 32;          // 3125, exact (2 waves/block)

    // layer 1
    aggregate_kernel<<<agg_blocks, 256, 0, stream>>>(feat, norm_src, src, dst, agg);
    gemm_bias_relu_kernel<<<gemm_blocks, 64, 0, stream>>>(agg, norm_dst, w1, b1, h1);

    // layer 2 (reuse agg)
    zero_kernel<<<(AGG_ELEMS + 255) / 256, 256, 0, stream>>>(agg, AGG_ELEMS);
    aggregate_kernel<<<agg_blocks, 256, 0, stream>>>(h1, norm_src, src, dst, agg);
    gemm_bias_relu_kernel<<<gemm_blocks, 64, 0, stream>>>(agg, norm_dst, w2, b2, hout);

    // gated pooling
    gate_pool_kernel<<<(N_NODES * 32) / 256, 256, 0, stream>>>(hout, wg, bg, gid, wh);
}